// Magnitude_19490561589307
// MI455X (gfx1250) — compile-verified
//
#include <hip/hip_runtime.h>
#include <math.h>
#include <float.h>

#define G_N   10000
#define L_N   1000
#define NSTA  400
#define NSRC  2000
#define K_N   15

typedef float v2f __attribute__((ext_vector_type(2)));
typedef float v8f __attribute__((ext_vector_type(8)));

__device__ __forceinline__ float softplus_f(float x) {
    // jax.nn.softplus: log1p(exp(-|x|)) + max(x, 0)
    return fmaxf(x, 0.0f) + log1pf(expf(-fabsf(x)));
}

// ---------------------------------------------------------------------------
// Kernel 1: sta_ind[n] = argmin_r ||sta[n] - locs_ref[r]||^2  (400 x 1000)
// Distance tiles via V_WMMA_F32_16X16X4_F32:
//   A row m = (sx, sy, sz, 1), B col n = (-2gx, -2gy, -2gz, |g|^2)
//   => C[m][n] = |g|^2 - 2 s.g   (true d^2 minus per-row const |s|^2,
//      which does not affect the per-row argmin)
// One wave handles 16 stations. Hot loop is just: B setup -> wmma ->
// 8 branchless per-lane running-min updates. The cross-lane (16-lane-half)
// argmin reduction happens ONCE after the tile loop.
// A layout (32-bit 16x4): lanes 0-15 -> {K0,K1}, lanes 16-31 -> {K2,K3}.
// C layout: VGPR v, lanes 0-15 -> row v, lanes 16-31 -> row v+8; col = lane&15.
// ---------------------------------------------------------------------------
__global__ void sta_ind_kernel(const float* __restrict__ sta,
                               const float* __restrict__ locs_ref,
                               int* __restrict__ sta_ind) {
    const int lane = threadIdx.x;      // 0..31
    const int half = lane >> 4;        // 0 or 1
    const int sub  = lane & 15;
    const int st   = blockIdx.x * 16 + sub;

    const float sx = sta[st * 3 + 0];
    const float sy = sta[st * 3 + 1];
    const float sz = sta[st * 3 + 2];
    v2f a;
    a.x = half ? sz   : sx;   // K=2 : K=0
    a.y = half ? 1.0f : sy;   // K=3 : K=1

    float runV[8];
    int   runI[8];
#pragma unroll
    for (int v = 0; v < 8; ++v) { runV[v] = FLT_MAX; runI[v] = 0; }

    const int ntiles = (L_N + 15) / 16;   // 63
    for (int t = 0; t < ntiles; ++t) {
        const int r = t * 16 + sub;       // column (ref) index owned by this lane
        const bool valid = (r < L_N);
        const int rc = valid ? r : (L_N - 1);   // clamp to keep loads in range
        const float rx = locs_ref[rc * 3 + 0];
        const float ry = locs_ref[rc * 3 + 1];
        const float rz = locs_ref[rc * 3 + 2];
        const float n2 = valid ? (rx * rx + ry * ry + rz * rz) : 1.0e30f;

        v2f b;                                      // branchless B setup
        b.x = half ? (-2.0f * rz) : (-2.0f * rx);   // K=2 : K=0
        b.y = half ? n2           : (-2.0f * ry);   // K=3 : K=1

        v8f c = {};
        c = __builtin_amdgcn_wmma_f32_16x16x4_f32(false, a, false, b,
                                                  (short)0, c, false, false);
#pragma unroll
        for (int v = 0; v < 8; ++v) {               // branchless running min
            const bool bt = c[v] < runV[v];
            runV[v] = bt ? c[v] : runV[v];
            runI[v] = bt ? r    : runI[v];
        }
    }

    // One cross-lane argmin per row (offsets < 16 stay inside the half-wave)
#pragma unroll
    for (int v = 0; v < 8; ++v) {
        float bv = runV[v];
        int   bi = runI[v];
#pragma unroll
        for (int off = 8; off >= 1; off >>= 1) {
            const float ov = __shfl_xor(bv, off);
            const int   oi = __shfl_xor(bi, off);
            const bool better = (ov < bv) | ((ov == bv) & (oi < bi));
            bv = better ? ov : bv;
            bi = better ? oi : bi;
        }
        runV[v] = bv;
        runI[v] = bi;
    }

    if (sub == 0) {
#pragma unroll
        for (int v = 0; v < 8; ++v) {
            const int row = v + 8 * half;
            sta_ind[blockIdx.x * 16 + row] = runI[v];
        }
    }
}

// ---------------------------------------------------------------------------
// Kernel 2: idx[q, 0..14] = 15 nearest grid points to src[q] (out of 10000).
// One wave per source. Each lane keeps a sorted (ascending) top-15 list in
// registers over its strided slice, then 15 rounds of wave-wide min-extraction
// merge via xor shuffles. (Set membership only; downstream sum is order-free.)
// ---------------------------------------------------------------------------
__global__ void topk_kernel(const float* __restrict__ src,
                            const float* __restrict__ x_grid,
                            int* __restrict__ idx_out) {
    const int q    = blockIdx.x;
    const int lane = threadIdx.x;
    const float sx = src[q * 3 + 0];
    const float sy = src[q * 3 + 1];
    const float sz = src[q * 3 + 2];

    float tv[K_N];
    int   tg[K_N];
#pragma unroll
    for (int j = 0; j < K_N; ++j) { tv[j] = FLT_MAX; tg[j] = -1; }

    for (int gi = lane; gi < G_N; gi += 32) {
        const float3 g3 = *(const float3*)(x_grid + gi * 3);
        const float dx = g3.x - sx;
        const float dy = g3.y - sy;
        const float dz = g3.z - sz;
        const float d  = dx * dx + dy * dy + dz * dz;
        if (d < tv[K_N - 1]) {
            // branchless ripple insert into sorted ascending list
#pragma unroll
            for (int j = K_N - 1; j >= 1; --j) {
                const bool cp = d < tv[j - 1];
                const bool cj = d < tv[j];
                const float nv = cp ? tv[j - 1] : (cj ? d  : tv[j]);
                const int   ng = cp ? tg[j - 1] : (cj ? gi : tg[j]);
                tv[j] = nv; tg[j] = ng;
            }
            if (d < tv[0]) { tv[0] = d; tg[0] = gi; }
        }
    }

    // Merge: 15 rounds, extract global min of the 32 lane-local heads.
    for (int r = 0; r < K_N; ++r) {
        float bv = tv[0];
        int   bg = tg[0];
#pragma unroll
        for (int off = 16; off >= 1; off >>= 1) {
            const float ov = __shfl_xor(bv, off);
            const int   og = __shfl_xor(bg, off);
            const bool better = (ov < bv) | ((ov == bv) & (og < bg));
            bv = better ? ov : bv;
            bg = better ? og : bg;
        }
        // grid indices are uniquely partitioned per lane -> unique owner pops
        if (tg[0] == bg) {
#pragma unroll
            for (int j = 0; j < K_N - 1; ++j) { tv[j] = tv[j + 1]; tg[j] = tg[j + 1]; }
            tv[K_N - 1] = FLT_MAX; tg[K_N - 1] = -1;
        }
        if (lane == 0) idx_out[q * K_N + r] = bg;
    }
}

// ---------------------------------------------------------------------------
// Kernel 3: log_amp[q, n]. One block per source, threads stride over stations.
// ---------------------------------------------------------------------------
__global__ void magnitude_kernel(const float* __restrict__ sta,
                                 const float* __restrict__ src,
                                 const float* __restrict__ mag,
                                 const int*   __restrict__ phase,
                                 const float* __restrict__ x_grid,
                                 const float* __restrict__ mag_coef,
                                 const float* __restrict__ epi_coef,
                                 const float* __restrict__ dep_coef,
                                 const float* __restrict__ coefs,
                                 const float* __restrict__ coefs_ker,
                                 const int*   __restrict__ sta_ind,
                                 const int*   __restrict__ idx_ws,
                                 float* __restrict__ out) {
    const int q = blockIdx.x;
    __shared__ float sdx[K_N], sdy[K_N], sdz[K_N];
    __shared__ int   sg[K_N];

    const float qx = src[q * 3 + 0];
    const float qy = src[q * 3 + 1];
    const float qz = src[q * 3 + 2];

    if (threadIdx.x < K_N) {
        const int g = idx_ws[q * K_N + threadIdx.x];
        sg[threadIdx.x]  = g;
        const float3 g3 = *(const float3*)(x_grid + g * 3);
        sdx[threadIdx.x] = g3.x - qx;   // pos1[idx] - pos2
        sdy[threadIdx.x] = g3.y - qy;
        sdz[threadIdx.x] = g3.z - qz;
    }
    __syncthreads();

    const float mq  = mag[q];
    const float mc0 = softplus_f(mag_coef[0]), mc1 = softplus_f(mag_coef[1]);
    const float ec0 = softplus_f(epi_coef[0]), ec1 = softplus_f(epi_coef[1]);
    const float dc0 = dep_coef[0],             dc1 = dep_coef[1];

    for (int n = threadIdx.x; n < NSTA; n += blockDim.x) {
        const float stx = sta[n * 3 + 0];
        const float sty = sta[n * 3 + 1];
        const float stz = sta[n * 3 + 2];
        const int ph = phase[n];
        const int s  = sta_ind[n];

        float wsum = 0.0f, wb = 0.0f;
#pragma unroll
        for (int k = 0; k < K_N; ++k) {
            const int base = sg[k] * L_N + s;             // < 1e7, fits int
            const float3 ke = *(const float3*)(coefs_ker + base * 3);
            const float e0 = softplus_f(ke.x);
            const float e1 = softplus_f(ke.y);
            const float e2 = softplus_f(ke.z);
            const float t0 = sdx[k] / e0;
            const float t1 = sdy[k] / e1;
            const float t2 = sdz[k] / e2;
            const float w = expf(-0.5f * (t0 * t0 + t1 * t1 + t2 * t2));
            wsum += w;
            wb   += w * coefs[base * 2 + ph];
        }
        const float bias = wb / (wsum == 0.0f ? 1.0f : wsum);

        const float hx   = 1000.0f * (qx - stx);
        const float hy   = 1000.0f * (qy - sty);
        const float pld0 = log10f(sqrtf(hx * hx + hy * hy) + 1.0f);
        const float pldd = log10f(fabsf(qz - stz) + 1.0f);

        const float mc = ph ? mc1 : mc0;
        const float ec = ph ? ec1 : ec0;
        const float dc = ph ? dc1 : dc0;
        out[q * NSTA + n] = mq * mc - ec * pld0 + dc * pldd + bias;
    }
}

extern "C" void kernel_launch(void* const* d_in, const int* in_sizes, int n_in,
                              void* d_out, int out_size, void* d_ws, size_t ws_size,
                              hipStream_t stream) {
    const float* sta       = (const float*)d_in[0];
    const float* src       = (const float*)d_in[1];
    const float* mag       = (const float*)d_in[2];
    const int*   phase     = (const int*)  d_in[3];
    const float* x_grid    = (const float*)d_in[4];
    const float* locs_ref  = (const float*)d_in[5];
    const float* mag_coef  = (const float*)d_in[6];
    const float* epi_coef  = (const float*)d_in[7];
    const float* dep_coef  = (const float*)d_in[8];
    const float* coefs     = (const float*)d_in[9];
    const float* coefs_ker = (const float*)d_in[10];
    float* out = (float*)d_out;

    int* sta_ind = (int*)d_ws;                         // 400 ints
    int* idxbuf  = (int*)((char*)d_ws + 2048);         // 2000*15 ints

    sta_ind_kernel<<<NSTA / 16, 32, 0, stream>>>(sta, locs_ref, sta_ind);
    topk_kernel<<<NSRC, 32, 0, stream>>>(src, x_grid, idxbuf);
    magnitude_kernel<<<NSRC, 256, 0, stream>>>(sta, src, mag, phase, x_grid,
                                               mag_coef, epi_coef, dep_coef,
                                               coefs, coefs_ker, sta_ind, idxbuf, out);
}